// RelationDistillationLoss_19250043420782
// MI455X (gfx1250) — compile-verified
//
#include <hip/hip_runtime.h>
#include <math.h>

#define KCLS 19
#define CCH  256
#define HWSZ (128 * 128)
#define BIMG 8
#define NPIX (BIMG * HWSZ)          // 131072
#define RED_BLOCK 128               // 4 waves; 2 blocks co-resident per WGP
#define PIX_PER_THREAD 4
#define PIX_PER_BLOCK (RED_BLOCK * PIX_PER_THREAD)  // 512 (divides HWSZ)
#define NBLK (NPIX / PIX_PER_BLOCK)                 // 256
#define ACC_STRIDE (CCH + 1)        // LDS bank spread across classes
#define MSTRIDE 258                 // pad so rows land in different banks

typedef __attribute__((ext_vector_type(2))) float v2f;
typedef __attribute__((ext_vector_type(8))) float v8f;
typedef __attribute__((ext_vector_type(4))) float v4f;   // true vector types
typedef __attribute__((ext_vector_type(4))) int   v4i;   // for NT loads

// ---------------------------------------------------------------- zero ws
__global__ void zero_ws(float* ws, int n) {
  int i = blockIdx.x * blockDim.x + threadIdx.x;
  if (i < n) ws[i] = 0.f;
}

// ------------------------------------------------- segment-sum reduction
// Grid: NBLK blocks x 128 threads. Each thread handles 4 consecutive
// pixels; features streamed with non-temporal float4 loads (read-once,
// working set > L2), scattered into padded LDS accumulators with
// ds_add_f32, flushed once per block with global atomic f32 adds.
__global__ __launch_bounds__(RED_BLOCK) void seg_reduce(
    const float* __restrict__ fs, const float* __restrict__ ft,
    const int* __restrict__ labels,
    float* __restrict__ gS, float* __restrict__ gT, float* __restrict__ gC) {
  __shared__ float accS[KCLS][ACC_STRIDE];
  __shared__ float accT[KCLS][ACC_STRIDE];
  __shared__ float cnt[KCLS];

  const int tid = threadIdx.x;
  for (int c = tid; c < CCH; c += RED_BLOCK) {
    for (int k = 0; k < KCLS; ++k) {
      accS[k][c] = 0.f;
      accT[k][c] = 0.f;
    }
  }
  if (tid < KCLS) cnt[tid] = 0.f;
  __syncthreads();

  const int base = blockIdx.x * PIX_PER_BLOCK;  // lies fully inside one image
  const int b = base / HWSZ;
  const int hw = (base % HWSZ) + tid * PIX_PER_THREAD;

  const v4i lab4 =
      __builtin_nontemporal_load((const v4i*)(labels + base + tid * PIX_PER_THREAD));
  const int k0 = lab4.x, k1 = lab4.y, k2 = lab4.z, k3 = lab4.w;
  atomicAdd(&cnt[k0], 1.f);
  atomicAdd(&cnt[k1], 1.f);
  atomicAdd(&cnt[k2], 1.f);
  atomicAdd(&cnt[k3], 1.f);

  const float* ps = fs + (size_t)b * CCH * HWSZ + hw;
  const float* pt = ft + (size_t)b * CCH * HWSZ + hw;
  for (int c = 0; c < CCH; ++c) {
    const v4f vs = __builtin_nontemporal_load((const v4f*)(ps + (size_t)c * HWSZ));
    const v4f vt = __builtin_nontemporal_load((const v4f*)(pt + (size_t)c * HWSZ));
    atomicAdd(&accS[k0][c], vs.x);
    atomicAdd(&accS[k1][c], vs.y);
    atomicAdd(&accS[k2][c], vs.z);
    atomicAdd(&accS[k3][c], vs.w);
    atomicAdd(&accT[k0][c], vt.x);
    atomicAdd(&accT[k1][c], vt.y);
    atomicAdd(&accT[k2][c], vt.z);
    atomicAdd(&accT[k3][c], vt.w);
  }
  __syncthreads();

  for (int c = tid; c < CCH; c += RED_BLOCK) {
    for (int k = 0; k < KCLS; ++k) {
      atomicAdd(&gS[k * CCH + c], accS[k][c]);
      atomicAdd(&gT[k * CCH + c], accT[k][c]);
    }
  }
  if (tid < KCLS) atomicAdd(&gC[tid], cnt[tid]);
}

// --------------------------------------- finalize: normalize, WMMA gram,
// off-diagonal softmax KD loss. Single block, 256 threads (8 waves).
__global__ __launch_bounds__(256) void finalize(
    const float* __restrict__ gS, const float* __restrict__ gT,
    const float* __restrict__ gC, float* __restrict__ out) {
  __shared__ float M[32][MSTRIDE];   // padded unit-row class means
  __shared__ float G[2][32][32];     // gram matrices (s, t)
  __shared__ float red[KCLS];        // per-class sum of squares

  const int tid = threadIdx.x;
  const int lane = tid & 31;
  const int wid = tid >> 5;

  for (int src = 0; src < 2; ++src) {
    const float* gsum = src ? gT : gS;

    // mean = sum / (count * C); absent classes stay zero rows.
    for (int k = 0; k < KCLS; ++k) {
      const float c = gC[k];
      M[k][tid] = (c > 0.f) ? gsum[k * CCH + tid] / (c * (float)CCH) : 0.f;
    }
    for (int k = KCLS; k < 32; ++k) M[k][tid] = 0.f;
    if (tid < KCLS) red[tid] = 0.f;
    __syncthreads();

    for (int k = 0; k < KCLS; ++k) {
      const float m = M[k][tid];
      atomicAdd(&red[k], m * m);
    }
    __syncthreads();

    for (int k = 0; k < KCLS; ++k) {
      const float nrm = sqrtf(red[k]);
      M[k][tid] *= 1.f / fmaxf(nrm, 1e-12f);
    }
    __syncthreads();

    // 4 waves, one 16x16 tile each of G = M * M^T (32x32 padded).
    // A and B fragments of V_WMMA_F32_16X16X4_F32 share the same per-lane
    // layout for M*M^T: lane (l2,hi) holds (M[r0+l2][k+2hi], M[r0+l2][k+2hi+1]).
    if (wid < 4) {
      const int i0 = (wid >> 1) * 16;
      const int j0 = (wid & 1) * 16;
      const int l2 = lane & 15;
      const int hi = lane >> 4;
      v8f acc = {};
      for (int k4 = 0; k4 < CCH; k4 += 4) {
        v2f a, bf;
        a.x = M[i0 + l2][k4 + 2 * hi];
        a.y = M[i0 + l2][k4 + 2 * hi + 1];
        bf.x = M[j0 + l2][k4 + 2 * hi];
        bf.y = M[j0 + l2][k4 + 2 * hi + 1];
        acc = __builtin_amdgcn_wmma_f32_16x16x4_f32(
            false, a, false, bf, (short)0, acc, false, false);
      }
      // D layout: lane<16 -> row v, col lane; lane>=16 -> row v+8, col lane-16
      for (int v = 0; v < 8; ++v) {
        G[src][i0 + v + (hi ? 8 : 0)][j0 + l2] = acc[v];
      }
    }
    __syncthreads();  // also protects M before next src overwrites it
  }

  if (tid == 0) {
    const float invT = 5.0f;  // 1 / 0.2
    float loss = 0.f;
    for (int i = 0; i < KCLS; ++i) {
      float maxs = -3.4e38f, maxt = -3.4e38f;
      for (int j = 0; j < KCLS; ++j) {
        if (j == i) continue;
        maxs = fmaxf(maxs, G[0][i][j] * invT);
        maxt = fmaxf(maxt, G[1][i][j] * invT);
      }
      float es = 0.f, et = 0.f;
      for (int j = 0; j < KCLS; ++j) {
        if (j == i) continue;
        es += expf(G[0][i][j] * invT - maxs);
        et += expf(G[1][i][j] * invT - maxt);
      }
      const float lses = maxs + logf(es);
      const float lset = maxt + logf(et);
      float row = 0.f;
      for (int j = 0; j < KCLS; ++j) {
        if (j == i) continue;
        const float logps = G[0][i][j] * invT - lses;
        const float pt = expf(G[1][i][j] * invT - lset);
        row += -pt * logps;
      }
      loss += row;
    }
    out[0] = loss / (float)KCLS;
  }
}

// ----------------------------------------------------------------- launch
extern "C" void kernel_launch(void* const* d_in, const int* in_sizes, int n_in,
                              void* d_out, int out_size, void* d_ws,
                              size_t ws_size, hipStream_t stream) {
  const float* fs = (const float*)d_in[0];
  const float* ft = (const float*)d_in[1];
  const int* labels = (const int*)d_in[2];

  float* ws = (float*)d_ws;
  float* gS = ws;
  float* gT = ws + KCLS * CCH;
  float* gC = ws + 2 * KCLS * CCH;
  const int nws = 2 * KCLS * CCH + KCLS;

  zero_ws<<<(nws + 255) / 256, 256, 0, stream>>>(ws, nws);
  seg_reduce<<<NBLK, RED_BLOCK, 0, stream>>>(fs, ft, labels, gS, gT, gC);
  finalize<<<1, 256, 0, stream>>>(gS, gT, gC, (float*)d_out);
}